// ConvTemporalGraphical_65249143161136
// MI455X (gfx1250) — compile-verified
//
#include <hip/hip_runtime.h>
#include <hip/hip_bf16.h>

typedef __attribute__((ext_vector_type(2))) float v2f;
typedef __attribute__((ext_vector_type(8))) float v8f;

#define NN   128
#define CIN  64
#define TT   12
#define VV   120
#define COUT 64
#define PP   (VV * VV)            /* 14400 positions per n      */
#define TV   (TT * VV)            /* 1440  (t,v) pairs per n    */
#define OUT0 ((size_t)NN * COUT * TT * VV)        /* 11,796,480 */
#define A2HALF ((size_t)NN * COUT * PP)           /* 117,964,800 */

static __device__ __forceinline__ v8f wmma_f32(v2f a, v2f b, v8f c) {
  return __builtin_amdgcn_wmma_f32_16x16x4_f32(false, a, false, b, (short)0, c,
                                               false, false);
}

// ---------------------------------------------------------------------------
// Kernel 1: xc[n, o, t, v] = sum_i Wc[o,i] * x[n,i,t,v] + bc[o]
// One wave = one 16x16 tile: rows = 16 output channels, cols = 16 (t,v) pairs.
// Block of 8 waves covers all 8 o-tiles for one (n, p-tile).
// ---------------------------------------------------------------------------
__global__ __launch_bounds__(256) void k_pointwise(
    const float* __restrict__ x, const float* __restrict__ Wc,
    const float* __restrict__ bc, float* __restrict__ xc) {
  const int lane = threadIdx.x & 31;
  const int wave = threadIdx.x >> 5;
  const int col  = lane & 15;
  const int hi   = lane >> 4;
  const int blk  = blockIdx.x;           // n * 90 + ptile
  const int n    = blk / 90;
  const int p0   = (blk % 90) * 16;
  const int o0   = wave * 16;

  const float* __restrict__ xb = x + (size_t)n * CIN * TV;

  v8f acc;
#pragma unroll
  for (int g = 0; g < 8; ++g) acc[g] = bc[o0 + g + 8 * hi];

#pragma unroll 4
  for (int k = 0; k < CIN; k += 4) {
    const int kk = k + 2 * hi;
    v2f a, b;
    a.x = Wc[(o0 + col) * CIN + kk];
    a.y = Wc[(o0 + col) * CIN + kk + 1];
    b.x = xb[(size_t)kk * TV + p0 + col];
    b.y = xb[(size_t)(kk + 1) * TV + p0 + col];
    acc = wmma_f32(a, b, acc);
  }

  float* __restrict__ dst = xc + ((size_t)n * 2 * COUT + o0) * TV;
#pragma unroll
  for (int g = 0; g < 8; ++g)
    dst[(size_t)(g + 8 * hi) * TV + p0 + col] = acc[g];
}

// ---------------------------------------------------------------------------
// Kernel 2: adj = W3·relu(W2·relu(W1·A[:, :5])) at every (n,v,w), then
// A2[0] = adj * (A[:,5] with zeroed diagonal), A2[1] = adj * eye.
// Layer 1 (K=5) is computed per-lane directly in the B-register layout the
// layer-2 WMMA needs.  Layer-2 D tiles go through LDS ([pos][ch]) to become
// layer-3 B operands.  One wave = 16 positions; block = 8 waves.
// ---------------------------------------------------------------------------
__global__ __launch_bounds__(256) void k_adj(
    const float* __restrict__ A,
    const float* __restrict__ W1, const float* __restrict__ b1,
    const float* __restrict__ W2, const float* __restrict__ b2,
    const float* __restrict__ W3, const float* __restrict__ b3,
    float* __restrict__ A2_0, float* __restrict__ A2_1) {
  __shared__ float lds[8 * 16 * 32];     // per-wave 16 pos x 32 ch
  const int lane = threadIdx.x & 31;
  const int wave = threadIdx.x >> 5;
  const int col  = lane & 15;
  const int hi   = lane >> 4;
  const int tile = blockIdx.x * 8 + wave;      // 0 .. 115199
  const int n    = tile / 900;
  const int p0   = (tile % 900) * 16;
  const int pos  = p0 + col;                   // 0 .. 14399
  const int vrow = pos / VV;
  const int wcol = pos - vrow * VV;

  const float* __restrict__ Ab = A + (size_t)n * 6 * PP;
  float ain[5];
#pragma unroll
  for (int i = 0; i < 5; ++i) ain[i] = Ab[(size_t)i * PP + pos];
  const float m5   = Ab[(size_t)5 * PP + pos];
  const float offd = (vrow == wcol) ? 0.0f : m5;
  const float diag = (vrow == wcol) ? 1.0f : 0.0f;

  // ---- layer 1 (VALU, K=5), produced in B layout for layer-2 WMMA --------
  v2f h1[4];
#pragma unroll
  for (int j = 0; j < 4; ++j) {
    const int c0 = 4 * j + 2 * hi;
    float s0 = b1[c0], s1 = b1[c0 + 1];
#pragma unroll
    for (int i = 0; i < 5; ++i) {
      s0 = fmaf(W1[c0 * 5 + i], ain[i], s0);
      s1 = fmaf(W1[(c0 + 1) * 5 + i], ain[i], s1);
    }
    h1[j].x = fmaxf(s0, 0.0f);
    h1[j].y = fmaxf(s1, 0.0f);
  }

  // ---- layer 2 (WMMA, M=32 -> 2 tiles, K=16) -----------------------------
  float* __restrict__ L = lds + wave * 512;
#pragma unroll
  for (int mt = 0; mt < 2; ++mt) {
    v8f acc;
#pragma unroll
    for (int g = 0; g < 8; ++g) acc[g] = b2[16 * mt + g + 8 * hi];
#pragma unroll
    for (int j = 0; j < 4; ++j) {
      const int kk = 4 * j + 2 * hi;
      v2f a;
      a.x = W2[(16 * mt + col) * 16 + kk];
      a.y = W2[(16 * mt + col) * 16 + kk + 1];
      acc = wmma_f32(a, h1[j], acc);
    }
#pragma unroll
    for (int g = 0; g < 8; ++g)                // relu + D->LDS ([pos][ch])
      L[col * 32 + 16 * mt + 8 * hi + g] = fmaxf(acc[g], 0.0f);
  }
  __syncthreads();

  v2f h2[8];
#pragma unroll
  for (int j = 0; j < 8; ++j) {                // reload in B layout
    const int kk = 4 * j + 2 * hi;
    h2[j].x = L[col * 32 + kk];
    h2[j].y = L[col * 32 + kk + 1];
  }

  // ---- layer 3 (WMMA, M=64 -> 4 tiles, K=32), mask + store ---------------
#pragma unroll
  for (int mt = 0; mt < 4; ++mt) {
    v8f acc;
#pragma unroll
    for (int g = 0; g < 8; ++g) acc[g] = b3[16 * mt + g + 8 * hi];
#pragma unroll
    for (int j = 0; j < 8; ++j) {
      const int kk = 4 * j + 2 * hi;
      v2f a;
      a.x = W3[(16 * mt + col) * 32 + kk];
      a.y = W3[(16 * mt + col) * 32 + kk + 1];
      acc = wmma_f32(a, h2[j], acc);
    }
#pragma unroll
    for (int g = 0; g < 8; ++g) {
      const int c = 16 * mt + g + 8 * hi;
      const size_t idx = ((size_t)n * COUT + c) * PP + pos;
      A2_0[idx] = acc[g] * offd;
      A2_1[idx] = acc[g] * diag;
    }
  }
}

// ---------------------------------------------------------------------------
// Kernel 3: out[n,c,t,w] = relu( sum_v xc[n,c,t,v] * A2_0[n,c,v,w]
//                                + xc[n,64+c,t,w] * A2_1[n,c,w,w] )
// One wave = one (n, c, 16-wide w tile); rows = t (12 valid of 16).
// ---------------------------------------------------------------------------
__global__ __launch_bounds__(256) void k_final(
    const float* __restrict__ xc, const float* __restrict__ A2_0,
    const float* __restrict__ A2_1, float* __restrict__ out) {
  const int lane = threadIdx.x & 31;
  const int wave = threadIdx.x >> 5;
  const int col  = lane & 15;
  const int hi   = lane >> 4;
  const int tile = blockIdx.x * 8 + wave;      // 0 .. 65535
  const int wt   = tile & 7;
  const int c    = (tile >> 3) & 63;
  const int n    = tile >> 9;
  const int w0   = wt * 16;

  const float* __restrict__ X0 = xc + ((size_t)n * 2 * COUT + c) * TV;
  const float* __restrict__ X1 = xc + ((size_t)n * 2 * COUT + COUT + c) * TV;
  const float* __restrict__ M  = A2_0 + ((size_t)n * COUT + c) * PP;
  const float* __restrict__ Dg = A2_1 + ((size_t)n * COUT + c) * PP;

  v8f acc;
#pragma unroll
  for (int g = 0; g < 8; ++g) acc[g] = 0.0f;

#pragma unroll 5
  for (int k = 0; k < VV; k += 4) {
    const int kk = k + 2 * hi;
    v2f a, b;
    a.x = (col < TT) ? X0[col * VV + kk] : 0.0f;
    a.y = (col < TT) ? X0[col * VV + kk + 1] : 0.0f;
    b.x = M[(size_t)kk * VV + w0 + col];
    b.y = M[(size_t)(kk + 1) * VV + w0 + col];
    acc = wmma_f32(a, b, acc);
  }

  const int w = w0 + col;
  const float dg = (w < VV) ? Dg[(size_t)w * (VV + 1)] : 0.0f;
#pragma unroll
  for (int g = 0; g < 8; ++g) {
    const int t = g + 8 * hi;
    if (t < TT && w < VV) {
      const float val = fmaf(X1[t * VV + w], dg, acc[g]);
      out[((size_t)(n * COUT + c) * TT + t) * VV + w] = fmaxf(val, 0.0f);
    }
  }
}

// ---------------------------------------------------------------------------
extern "C" void kernel_launch(void* const* d_in, const int* in_sizes, int n_in,
                              void* d_out, int out_size, void* d_ws,
                              size_t ws_size, hipStream_t stream) {
  const float* x  = (const float*)d_in[0];
  const float* A  = (const float*)d_in[1];
  const float* Wc = (const float*)d_in[2];
  const float* bc = (const float*)d_in[3];
  const float* W1 = (const float*)d_in[4];
  const float* b1 = (const float*)d_in[5];
  const float* W2 = (const float*)d_in[6];
  const float* b2 = (const float*)d_in[7];
  const float* W3 = (const float*)d_in[8];
  const float* b3 = (const float*)d_in[9];

  float* out  = (float*)d_out;
  float* A2_0 = out + OUT0;            // [0, n, 64, v, w]
  float* A2_1 = A2_0 + A2HALF;         // [1, n, 64, v, w]
  float* xc   = (float*)d_ws;          // [n, 128, t, v]  (94.4 MB)

  k_pointwise<<<NN * 90, 256, 0, stream>>>(x, Wc, bc, xc);
  k_adj<<<(NN * 900) / 8, 256, 0, stream>>>(A, W1, b1, W2, b2, W3, b3,
                                            A2_0, A2_1);
  k_final<<<(NN * COUT * 8) / 8, 256, 0, stream>>>(xc, A2_0, A2_1, out);
}